// ACRD_20718922235949
// MI455X (gfx1250) — compile-verified
//
#include <hip/hip_runtime.h>
#include <math.h>

// Problem constants (fixed by reference setup_inputs)
#define BATCH 256
#define DIM   2048
#define N0    7936
#define NGAL  8192   // N0 + BATCH
#define KTOP  128
#define EPSV  1e-12f

typedef float v2f __attribute__((ext_vector_type(2)));
typedef float v8f __attribute__((ext_vector_type(8)));

// ---------------------------------------------------------------------------
// 1) Elementwise squares: fs = f_s^2, ft = f_t^2
// ---------------------------------------------------------------------------
__global__ void square_kernel(const float* __restrict__ f_s,
                              const float* __restrict__ f_t,
                              float* __restrict__ fs,
                              float* __restrict__ ft) {
    int i = blockIdx.x * blockDim.x + threadIdx.x;   // grid sized exactly
    float a = f_s[i];
    float c = f_t[i];
    fs[i] = a * a;
    ft[i] = c * c;
}

// ---------------------------------------------------------------------------
// 2) Row sum-of-squares. Blocks 0..N0-1: gallery rows -> galnorm,
//    N0..NGAL-1: ft rows -> galnorm, NGAL..NGAL+BATCH-1: fs rows -> fsnorm.
// ---------------------------------------------------------------------------
__global__ void norms_kernel(const float* __restrict__ gallery,
                             const float* __restrict__ ft,
                             const float* __restrict__ fs,
                             float* __restrict__ galnorm,
                             float* __restrict__ fsnorm) {
    int i = blockIdx.x;
    const float* row;
    float* dst;
    if (i < N0)          { row = gallery + (size_t)i * DIM;          dst = galnorm + i; }
    else if (i < NGAL)   { row = ft + (size_t)(i - N0) * DIM;        dst = galnorm + i; }
    else                 { row = fs + (size_t)(i - NGAL) * DIM;      dst = fsnorm + (i - NGAL); }

    int tid = threadIdx.x;
    float s = 0.f;
    for (int j = tid; j < DIM; j += 256) { float v = row[j]; s += v * v; }
    for (int off = 16; off; off >>= 1) s += __shfl_xor(s, off, 32);
    __shared__ float sred[8];
    if ((tid & 31) == 0) sred[tid >> 5] = s;
    __syncthreads();
    if (tid == 0) {
        float tot = 0.f;
        #pragma unroll
        for (int w = 0; w < 8; ++w) tot += sred[w];
        *dst = tot;
    }
}

// ---------------------------------------------------------------------------
// 3) WMMA fp32 GEMM + fused distance epilogue.
//    D = ft @ gal^T  (256 x 8192, K = 2048), tt = sqrt(max(na+nb-2D, eps)).
//    One wave -> 16(M) x 64(N) strip via 4 accumulator tiles of 16x16.
//    V_WMMA_F32_16X16X4_F32 operand layout (32-bit A 16x4 / B 4x16):
//      lane L, row = (L&15): float2 at [row][kbase + 2*(L>>4)].
// ---------------------------------------------------------------------------
__global__ void gemm_dist_kernel(const float* __restrict__ ft,
                                 const float* __restrict__ gallery,
                                 const float* __restrict__ galnorm,
                                 float* __restrict__ ttd) {
    int tid    = threadIdx.x;
    int lane   = tid & 31;
    int waveId = tid >> 5;
    int w      = blockIdx.x * 8 + waveId;       // 0 .. 2047
    int tileM  = (w >> 7) << 4;                 // 0,16,...,240
    int tileN  = (w & 127) << 6;                // 0,64,...,8128

    int half = lane >> 4;                       // 0 or 1
    int lrow = lane & 15;
    int kOff = half * 2;

    const float* aRow = ft + (size_t)(tileM + lrow) * DIM + kOff;
    const float* bRow[4];
    #pragma unroll
    for (int s = 0; s < 4; ++s) {
        int n = tileN + s * 16 + lrow;
        bRow[s] = (n < N0 ? gallery + (size_t)n * DIM
                          : ft + (size_t)(n - N0) * DIM) + kOff;
    }

    v8f zero = {0.f, 0.f, 0.f, 0.f, 0.f, 0.f, 0.f, 0.f};
    v8f acc[4] = {zero, zero, zero, zero};

    for (int kk = 0; kk < DIM; kk += 64) {
        // prefetch next chunk of A and B rows into cache
        __builtin_prefetch(aRow + kk + 64, 0, 1);
        #pragma unroll
        for (int s = 0; s < 4; ++s) __builtin_prefetch(bRow[s] + kk + 64, 0, 1);

        #pragma unroll
        for (int k2 = 0; k2 < 64; k2 += 4) {
            int k = kk + k2;
            v2f a = *(const v2f*)(aRow + k);
            #pragma unroll
            for (int s = 0; s < 4; ++s) {
                v2f b = *(const v2f*)(bRow[s] + k);
                acc[s] = __builtin_amdgcn_wmma_f32_16x16x4_f32(
                    /*neg_a=*/false, a, /*neg_b=*/false, b,
                    /*c_mod=*/(short)0, acc[s],
                    /*reuse_a=*/false, /*reuse_b=*/false);
            }
        }
    }

    // epilogue: d2 = ||ft_m||^2 + ||gal_n||^2 - 2*dot ; tt = sqrt(clamp(d2, eps))
    #pragma unroll
    for (int s = 0; s < 4; ++s) {
        int n = tileN + s * 16 + lrow;
        float gn = galnorm[n];
        #pragma unroll
        for (int r = 0; r < 8; ++r) {
            int m = tileM + r + half * 8;
            float d2 = galnorm[N0 + m] + gn - 2.0f * acc[s][r];
            ttd[(size_t)m * NGAL + n] = sqrtf(fmaxf(d2, EPSV));
        }
    }
}

// ---------------------------------------------------------------------------
// 4) Per-row top-K smallest (ascending) with indices. One block per row.
//    8192 values live in LDS (32KB << 320KB/WGP); 128 iterative argmins.
// ---------------------------------------------------------------------------
__global__ void topk_kernel(const float* __restrict__ ttd,
                            float* __restrict__ tval,
                            int* __restrict__ tidx) {
    __shared__ float vals[NGAL];
    __shared__ float rv[256];
    __shared__ int   ri[256];
    int b = blockIdx.x;
    int tid = threadIdx.x;
    const float* row = ttd + (size_t)b * NGAL;
    for (int j = tid; j < NGAL; j += 256) vals[j] = row[j];
    __syncthreads();

    for (int t = 0; t < KTOP; ++t) {
        float mv = 3.4e38f; int mi = 0;
        for (int j = tid; j < NGAL; j += 256) {
            float v = vals[j];
            if (v < mv) { mv = v; mi = j; }
        }
        rv[tid] = mv; ri[tid] = mi;
        __syncthreads();
        for (int s = 128; s > 0; s >>= 1) {
            if (tid < s) {
                if (rv[tid + s] < rv[tid]) { rv[tid] = rv[tid + s]; ri[tid] = ri[tid + s]; }
            }
            __syncthreads();
        }
        if (tid == 0) {
            tval[b * KTOP + t] = rv[0];
            tidx[b * KTOP + t] = ri[0];
            vals[ri[0]] = 3.4e38f;     // remove winner
        }
        __syncthreads();
    }
}

// ---------------------------------------------------------------------------
// 5) Student distances to selected gallery rows. One wave per (b,k) pair.
// ---------------------------------------------------------------------------
__global__ void sdist_kernel(const float* __restrict__ fs,
                             const float* __restrict__ ft,
                             const float* __restrict__ gallery,
                             const float* __restrict__ galnorm,
                             const float* __restrict__ fsnorm,
                             const int* __restrict__ tidx,
                             float* __restrict__ sval) {
    int tid  = threadIdx.x;
    int lane = tid & 31;
    int w    = blockIdx.x * 8 + (tid >> 5);    // 0 .. 32767
    int b    = w >> 7;
    int idx  = tidx[w];

    const float* rowp = (idx < N0) ? gallery + (size_t)idx * DIM
                                   : ft + (size_t)(idx - N0) * DIM;
    const float* fsr = fs + (size_t)b * DIM;

    float dot = 0.f;
    for (int j = lane * 4; j < DIM; j += 128) {
        float4 a = *(const float4*)(fsr + j);
        float4 g = *(const float4*)(rowp + j);
        dot += a.x * g.x + a.y * g.y + a.z * g.z + a.w * g.w;
    }
    for (int off = 16; off; off >>= 1) dot += __shfl_xor(dot, off, 32);

    if (lane == 0) {
        float d2 = fsnorm[b] + galnorm[idx] - 2.0f * dot;
        sval[w] = sqrtf(fmaxf(d2, EPSV));
    }
}

// ---------------------------------------------------------------------------
// 6) Loss: smooth-rank L2 on tail 127 values + |s0 - t0|; atomics into acc.
// ---------------------------------------------------------------------------
__global__ void zero_acc_kernel(float* __restrict__ acc) {
    if (threadIdx.x < 2) acc[threadIdx.x] = 0.f;
}

__global__ void loss_kernel(const float* __restrict__ tval,
                            const float* __restrict__ sval,
                            float* __restrict__ acc) {
    __shared__ float tv[KTOP], sv[KTOP], red[KTOP];
    int b = blockIdx.x, tid = threadIdx.x;
    tv[tid] = tval[b * KTOP + tid];
    sv[tid] = sval[b * KTOP + tid];
    __syncthreads();

    float term = 0.f;
    if (tid >= 1) {
        float ti = tv[tid], si = sv[tid];
        float tr = 0.f, sr = 0.f;
        for (int j = 1; j < KTOP; ++j) {
            tr += fmaxf(ti - tv[j], 0.f);
            sr += fmaxf(si - sv[j], 0.f);
        }
        float dd = tr - sr;
        term = dd * dd;
    }
    red[tid] = term;
    __syncthreads();
    for (int s = 64; s > 0; s >>= 1) {
        if (tid < s) red[tid] += red[tid + s];
        __syncthreads();
    }
    if (tid == 0) {
        atomicAdd(&acc[1], sqrtf(red[0]));          // ||t_rank - s_rank||_2
        atomicAdd(&acc[0], fabsf(sv[0] - tv[0]));   // |s0 - t0|
    }
}

__global__ void finalize_kernel(const float* __restrict__ acc,
                                float* __restrict__ out) {
    // ALPHA * mean(|s0-t0|) + BETA * mean(norm)/r
    out[0] = (acc[0] / (float)BATCH) + ((acc[1] / (float)BATCH) / (float)(KTOP - 1));
}

// ---------------------------------------------------------------------------
// Launch
// ---------------------------------------------------------------------------
extern "C" void kernel_launch(void* const* d_in, const int* in_sizes, int n_in,
                              void* d_out, int out_size, void* d_ws, size_t ws_size,
                              hipStream_t stream) {
    const float* f_s     = (const float*)d_in[0];   // [256, 2048]
    const float* f_t     = (const float*)d_in[1];   // [256, 2048]
    const float* gallery = (const float*)d_in[2];   // [7936, 2048]
    float* out = (float*)d_out;

    float* ws = (float*)d_ws;
    float* fs      = ws;                       // 524288
    float* ft      = fs + BATCH * DIM;         // 524288
    float* galnorm = ft + BATCH * DIM;         // 8192
    float* fsnorm  = galnorm + NGAL;           // 256
    float* ttd     = fsnorm + BATCH;           // 2097152
    float* tval    = ttd + (size_t)BATCH * NGAL; // 32768
    float* sval    = tval + BATCH * KTOP;      // 32768
    int*   tidx    = (int*)(sval + BATCH * KTOP); // 32768 ints
    float* acc     = (float*)(tidx + BATCH * KTOP); // 2 floats

    zero_acc_kernel<<<1, 32, 0, stream>>>(acc);
    square_kernel<<<(BATCH * DIM) / 256, 256, 0, stream>>>(f_s, f_t, fs, ft);
    norms_kernel<<<NGAL + BATCH, 256, 0, stream>>>(gallery, ft, fs, galnorm, fsnorm);
    gemm_dist_kernel<<<256, 256, 0, stream>>>(ft, gallery, galnorm, ttd);
    topk_kernel<<<BATCH, 256, 0, stream>>>(ttd, tval, tidx);
    sdist_kernel<<<(BATCH * KTOP) / 8, 256, 0, stream>>>(fs, ft, gallery, galnorm,
                                                         fsnorm, tidx, sval);
    loss_kernel<<<BATCH, KTOP, 0, stream>>>(tval, sval, acc);
    finalize_kernel<<<1, 1, 0, stream>>>(acc, out);
}